// FaissIVFPQLTM_61804579389944
// MI455X (gfx1250) — compile-verified
//
#include <hip/hip_runtime.h>

typedef __attribute__((ext_vector_type(2))) float v2f;
typedef __attribute__((ext_vector_type(8))) float v8f;

#define B_Q    64
#define DIM    256
#define NLIST  4096
#define NPROBE 8
#define PQM    16
#define KPQ    256
#define DSUB   16
#define NDB    100000
#define DV     512
#define CAP    8192
#define BIGF   1e30f

// ---------------- norms: ||q||^2, ||c||^2, ||codeword||^2 ----------------
__global__ void k_norms(const float* __restrict__ Q, const float* __restrict__ C,
                        const float* __restrict__ CB,
                        float* __restrict__ qn, float* __restrict__ cn, float* __restrict__ cbn) {
  int t = blockIdx.x * blockDim.x + threadIdx.x;
  if (t < NLIST) {
    const float* r = C + (size_t)t * DIM; float s = 0.f;
    for (int d = 0; d < DIM; ++d) s += r[d] * r[d];
    cn[t] = s;
  } else if (t < NLIST + PQM * KPQ) {
    int i = t - NLIST;
    const float* r = CB + (size_t)i * DSUB; float s = 0.f;
    for (int d = 0; d < DSUB; ++d) s += r[d] * r[d];
    cbn[i] = s;
  } else if (t < NLIST + PQM * KPQ + B_Q) {
    int i = t - NLIST - PQM * KPQ;
    const float* r = Q + (size_t)i * DIM; float s = 0.f;
    for (int d = 0; d < DIM; ++d) s += r[d] * r[d];
    qn[i] = s;
  }
}

// ---------------- coarse: qc = ||q||^2 + ||c||^2 - 2 q.c via WMMA f32 16x16x4 ----------------
// one wave per 16x16 output tile; 4 M-tiles x 256 N-tiles = 1024 waves
__global__ void k_coarse(const float* __restrict__ Q, const float* __restrict__ C,
                         const float* __restrict__ qn, const float* __restrict__ cn,
                         float* __restrict__ qc) {
  int wave = (blockIdx.x * blockDim.x + threadIdx.x) >> 5;
  int lane = threadIdx.x & 31;
  int tm = wave >> 8;        // 0..3
  int tn = wave & 255;       // 0..255
  int ln = lane & 15;
  int half = lane >> 4;
  const float* qrow = Q + (size_t)(tm * 16 + ln) * DIM;  // A: row m = lane%16
  const float* crow = C + (size_t)(tn * 16 + ln) * DIM;  // B = C^T: col n = lane%16
  v8f acc = {};
  for (int k0 = 0; k0 < DIM; k0 += 4) {
    v2f a, b;
    a.x = qrow[k0 + 2 * half]; a.y = qrow[k0 + 2 * half + 1];
    b.x = crow[k0 + 2 * half]; b.y = crow[k0 + 2 * half + 1];
    acc = __builtin_amdgcn_wmma_f32_16x16x4_f32(false, a, false, b, (short)0, acc, false, false);
  }
  int ncol = tn * 16 + ln;
#pragma unroll
  for (int v = 0; v < 8; ++v) {
    int mrow = tm * 16 + v + 8 * half;
    qc[(size_t)mrow * NLIST + ncol] = qn[mrow] + cn[ncol] - 2.0f * acc[v];
  }
}

// ---------------- per-query top-8 lists + residuals + residual sub-norms ----------------
__global__ void k_top8(const float* __restrict__ Q, const float* __restrict__ C,
                       const float* __restrict__ qc, int* __restrict__ probes,
                       float* __restrict__ resid, float* __restrict__ rsnorm) {
  __shared__ float sv[NLIST];
  __shared__ float rv[256];
  __shared__ int   ri[256];
  __shared__ int   pr[NPROBE];
  int b = blockIdx.x, tid = threadIdx.x;
  for (int i = tid; i < NLIST; i += blockDim.x) sv[i] = qc[(size_t)b * NLIST + i];
  __syncthreads();
  for (int r = 0; r < NPROBE; ++r) {
    float mv = BIGF; int mi = 0;
    for (int i = tid; i < NLIST; i += blockDim.x)
      if (sv[i] < mv) { mv = sv[i]; mi = i; }
    rv[tid] = mv; ri[tid] = mi;
    __syncthreads();
    for (int s = blockDim.x / 2; s > 0; s >>= 1) {
      if (tid < s && rv[tid + s] < rv[tid]) { rv[tid] = rv[tid + s]; ri[tid] = ri[tid + s]; }
      __syncthreads();
    }
    if (tid == 0) { pr[r] = ri[0]; probes[b * NPROBE + r] = ri[0]; }
    __syncthreads();
    if (tid == 0) sv[pr[r]] = BIGF;
    __syncthreads();
  }
  for (int i = tid; i < NPROBE * DIM; i += blockDim.x) {
    int p = i / DIM, d = i - p * DIM;
    resid[(size_t)(b * NPROBE + p) * DIM + d] = Q[(size_t)b * DIM + d] - C[(size_t)pr[p] * DIM + d];
  }
  __syncthreads();
  for (int i = tid; i < NPROBE * PQM; i += blockDim.x) {
    int p = i / PQM, m = i - p * PQM;
    const float* rr = resid + (size_t)(b * NPROBE + p) * DIM + m * DSUB;
    float s = 0.f;
    for (int d = 0; d < DSUB; ++d) s += rr[d] * rr[d];
    rsnorm[(b * NPROBE + p) * PQM + m] = s;
  }
}

// ---------------- ADC tables via WMMA: T[r][m][k] = |rs|^2 + |cb|^2 - 2 rs.cb ----------------
// rows r = b*8+p (512); per m: A 512x16, B 16x256 -> 32x16 tiles; 16*512 = 8192 waves
__global__ void k_tables(const float* __restrict__ resid, const float* __restrict__ CB,
                         const float* __restrict__ rsnorm, const float* __restrict__ cbn,
                         float* __restrict__ tables) {
  int wave = (blockIdx.x * blockDim.x + threadIdx.x) >> 5;
  int lane = threadIdx.x & 31;
  int m   = wave >> 9;       // 0..15
  int rem = wave & 511;
  int tr  = rem >> 4;        // 0..31
  int tn  = rem & 15;        // 0..15
  int ln = lane & 15, half = lane >> 4;
  const float* arow = resid + (size_t)(tr * 16 + ln) * DIM + m * DSUB;
  const float* brow = CB + ((size_t)m * KPQ + tn * 16 + ln) * DSUB;
  v8f acc = {};
#pragma unroll
  for (int k0 = 0; k0 < DSUB; k0 += 4) {
    v2f a, b;
    a.x = arow[k0 + 2 * half]; a.y = arow[k0 + 2 * half + 1];
    b.x = brow[k0 + 2 * half]; b.y = brow[k0 + 2 * half + 1];
    acc = __builtin_amdgcn_wmma_f32_16x16x4_f32(false, a, false, b, (short)0, acc, false, false);
  }
  int n = tn * 16 + ln;
#pragma unroll
  for (int v = 0; v < 8; ++v) {
    int r = tr * 16 + v + 8 * half;
    tables[(size_t)r * (PQM * KPQ) + m * KPQ + n] =
        rsnorm[r * PQM + m] + cbn[m * KPQ + n] - 2.0f * acc[v];
  }
}

// ---------------- init candidate counters ----------------
__global__ void k_init(int* __restrict__ ccnt) {
  int t = threadIdx.x;
  if (t < B_Q) ccnt[t] = 0;
}

// ---------------- scan: match list_id against probes, ADC distance, compact ----------------
__global__ void k_scan(const int* __restrict__ list_ids, const int* __restrict__ codes,
                       const int* __restrict__ probes, const float* __restrict__ tables,
                       float* __restrict__ cdist, int* __restrict__ cidx, int* __restrict__ ccnt) {
  int b = blockIdx.y;
  int pr[NPROBE];
#pragma unroll
  for (int j = 0; j < NPROBE; ++j) pr[j] = probes[b * NPROBE + j];
  int stride = gridDim.x * blockDim.x;
  for (int n = blockIdx.x * blockDim.x + threadIdx.x; n < NDB; n += stride) {
    int lid = list_ids[n];
    int p = -1;
#pragma unroll
    for (int j = 0; j < NPROBE; ++j)
      if (lid == pr[j]) p = j;
    if (p >= 0) {
      const float* tb = tables + (size_t)(b * NPROBE + p) * (PQM * KPQ);
      const int* cd = codes + (size_t)n * PQM;
      float dsum = 0.f;
#pragma unroll
      for (int j = 0; j < PQM; ++j) dsum += tb[j * KPQ + cd[j]];
      int slot = atomicAdd(&ccnt[b], 1);
      if (slot < CAP) {
        cdist[(size_t)b * CAP + slot] = dsum;
        cidx[(size_t)b * CAP + slot] = n;
      }
    }
  }
}

// ---------------- top-k over candidates + softmax + weighted value sum ----------------
__global__ void k_finalize(const float* __restrict__ cdist, const int* __restrict__ cidx,
                           const int* __restrict__ ccnt, const float* __restrict__ values,
                           const int* __restrict__ topk_p, float* __restrict__ out) {
  __shared__ float sd[CAP];      // 32 KB
  __shared__ float rv[256];
  __shared__ int   ri[256];
  __shared__ float sel_d[64];
  __shared__ int   sel_i[64];
  __shared__ float wts[64];
  int b = blockIdx.x, tid = threadIdx.x;
  int K = *topk_p; if (K > 64) K = 64; if (K < 1) K = 1;
  int cnt = ccnt[b]; if (cnt > CAP) cnt = CAP;
  for (int i = tid; i < cnt; i += blockDim.x) sd[i] = cdist[(size_t)b * CAP + i];
  __syncthreads();
  for (int k = 0; k < K; ++k) {
    float mv = BIGF; int mi = -1;
    for (int i = tid; i < cnt; i += blockDim.x)
      if (sd[i] < mv) { mv = sd[i]; mi = i; }
    rv[tid] = mv; ri[tid] = mi;
    __syncthreads();
    for (int s = blockDim.x / 2; s > 0; s >>= 1) {
      if (tid < s && rv[tid + s] < rv[tid]) { rv[tid] = rv[tid + s]; ri[tid] = ri[tid + s]; }
      __syncthreads();
    }
    if (tid == 0) {
      int pos = ri[0];
      if (pos >= 0) { sel_d[k] = rv[0]; sel_i[k] = cidx[(size_t)b * CAP + pos]; sd[pos] = BIGF; }
      else          { sel_d[k] = BIGF;  sel_i[k] = 0; }
    }
    __syncthreads();
  }
  if (tid == 0) {
    float dmin = sel_d[0];
    for (int k = 1; k < K; ++k) dmin = fminf(dmin, sel_d[k]);
    float sum = 0.f;
    for (int k = 0; k < K; ++k) { float w = __expf(-(sel_d[k] - dmin)); wts[k] = w; sum += w; }
    float inv = 1.0f / sum;
    for (int k = 0; k < K; ++k) wts[k] *= inv;
  }
  __syncthreads();
  for (int v = tid; v < DV; v += blockDim.x) {
    float acc = 0.f;
    for (int k = 0; k < K; ++k) acc += wts[k] * values[(size_t)sel_i[k] * DV + v];
    out[(size_t)b * DV + v] = acc;
  }
}

extern "C" void kernel_launch(void* const* d_in, const int* in_sizes, int n_in,
                              void* d_out, int out_size, void* d_ws, size_t ws_size,
                              hipStream_t stream) {
  (void)in_sizes; (void)n_in; (void)out_size; (void)ws_size;
  const float* Q        = (const float*)d_in[0];
  const float* C        = (const float*)d_in[1];
  const float* CB       = (const float*)d_in[2];
  const int*   codes    = (const int*)d_in[3];
  const int*   list_ids = (const int*)d_in[4];
  const float* values   = (const float*)d_in[5];
  const int*   topk     = (const int*)d_in[6];
  float* out = (float*)d_out;

  char* w = (char*)d_ws;
  float* qn     = (float*)w; w += (size_t)B_Q * 4;
  float* cn     = (float*)w; w += (size_t)NLIST * 4;
  float* cbn    = (float*)w; w += (size_t)PQM * KPQ * 4;
  float* qc     = (float*)w; w += (size_t)B_Q * NLIST * 4;
  int*   probes = (int*)w;   w += (size_t)B_Q * NPROBE * 4;
  float* resid  = (float*)w; w += (size_t)B_Q * NPROBE * DIM * 4;
  float* rsn    = (float*)w; w += (size_t)B_Q * NPROBE * PQM * 4;
  float* tables = (float*)w; w += (size_t)B_Q * NPROBE * PQM * KPQ * 4;
  int*   ccnt   = (int*)w;   w += (size_t)B_Q * 4;
  float* cdist  = (float*)w; w += (size_t)B_Q * CAP * 4;
  int*   cidx   = (int*)w;   w += (size_t)B_Q * CAP * 4;

  // 1) norms
  k_norms<<<(NLIST + PQM * KPQ + B_Q + 255) / 256, 256, 0, stream>>>(Q, C, CB, qn, cn, cbn);
  // 2) coarse GEMM (WMMA): 1024 waves -> 128 blocks x 256
  k_coarse<<<128, 256, 0, stream>>>(Q, C, qn, cn, qc);
  // 3) per-query top-8 + residuals
  k_top8<<<B_Q, 256, 0, stream>>>(Q, C, qc, probes, resid, rsn);
  // 4) ADC tables (WMMA): 8192 waves -> 1024 blocks x 256
  k_tables<<<1024, 256, 0, stream>>>(resid, CB, rsn, cbn, tables);
  // 5) reset candidate counters
  k_init<<<1, 64, 0, stream>>>(ccnt);
  // 6) match + compact scan
  k_scan<<<dim3(128, B_Q), 256, 0, stream>>>(list_ids, codes, probes, tables, cdist, cidx, ccnt);
  // 7) top-k + softmax + weighted sum of values
  k_finalize<<<B_Q, 256, 0, stream>>>(cdist, cidx, ccnt, values, topk, out);
}